// TimeEmbedding_50861002719746
// MI455X (gfx1250) — compile-verified
//
#include <hip/hip_runtime.h>

// ---------------------------------------------------------------------------
// Reference: z = one_hot(elt, V) @ W.T + b  ==  z[i,e] = W[e, elt[i]] + b[e]
// B=4096, E=128, V=100000.  Pure gather + bias: ~2MB out, <=33MB scattered
// reads (W fits in the 192MB L2) -> ~1.5us at 23.3 TB/s.  A WMMA one-hot
// matmul would be ~105 GFLOP + huge operand traffic: >30x slower. So the
// CDNA5 features used here are the data-movement ones: TDM tensor_load_to_lds
// (async DMA of the bias row into LDS, TENSORcnt) + s_wait_tensorcnt, LDS
// b128 reads, and wave32-coalesced b128 stores (one wave == one batch row).
// ---------------------------------------------------------------------------

typedef unsigned int u32;
typedef unsigned long long u64;
typedef __attribute__((ext_vector_type(4))) u32 u32x4;
typedef __attribute__((ext_vector_type(8))) int i32x8;
typedef __attribute__((ext_vector_type(4))) int i32x4;

#define EMB 128
#define WAVES_PER_BLOCK 8
#define THREADS (32 * WAVES_PER_BLOCK)

__global__ __launch_bounds__(THREADS) void
emb_gather_kernel(const int* __restrict__ elt,
                  const float* __restrict__ W,
                  const float* __restrict__ bias,
                  float* __restrict__ out,
                  int batch, int vocab) {
    __shared__ float sbias[EMB];

    // ---- Stage bias[0:128] (512B) into LDS with the Tensor Data Mover. ----
    // Only wave 0 reaches this block (uniform branch), so exactly one TDM op
    // is issued per workgroup (TDM ignores EXEC; issues once per wave).
    if (threadIdx.x < 32) {
        u32 lds_off = (u32)(u64)(__attribute__((address_space(3))) const float*)sbias;
        u64 ga = (u64)(const void*)bias;

        // D# group 0 (128b): [1:0]=count=1, [63:32]=lds_addr,
        // [120:64]=global_addr, [127:126]=type=2 ("image").
        u32x4 g0;
        g0.x = 1u;                                    // count=1, user desc
        g0.y = lds_off;                               // lds_addr (bytes)
        g0.z = (u32)(ga & 0xFFFFFFFFu);               // global_addr[31:0]
        g0.w = (u32)((ga >> 32) & 0x01FFFFFFu)        // global_addr[56:32]
             | 0x80000000u;                           // type = 2
        // D# group 1 (256b): data_size=4B; tensor_dim0=128 @ [79:48];
        // tensor_dim1=1 @ [111:80]; tile_dim0=128 @ [127:112];
        // tile_dim1=1 @ [143:128]; tensor_dim0_stride=128 @ [207:160].
        i32x8 g1;
        g1[0] = 2 << 16;                // data_size = 2 -> 4 bytes; mask=0
        g1[1] = (EMB & 0xFFFF) << 16;   // tensor_dim0 low16
        g1[2] = 1 << 16;                // tensor_dim0 hi=0 | tensor_dim1=1
        g1[3] = (EMB & 0xFFFF) << 16;   // tile_dim0 = 128
        g1[4] = 1;                      // tile_dim1 = 1, tile_dim2 = 0
        g1[5] = EMB;                    // tensor_dim0_stride low32
        g1[6] = 0;
        g1[7] = 0;
        i32x4 gz4 = {0, 0, 0, 0};              // groups 2/3 unused (<=2D tensor)
        i32x8 gz8 = {0, 0, 0, 0, 0, 0, 0, 0};  // extra group (6-arg builtin form)

        __builtin_amdgcn_tensor_load_to_lds(g0, g1, gz4, gz4, gz8, 0);
        __builtin_amdgcn_s_wait_tensorcnt(0);  // s_wait_tensorcnt 0
    }
    __syncthreads();

    // ---- Gather: one wave32 per batch row; lane q -> elements 4q..4q+3 ----
    const int lane = threadIdx.x & 31;
    const int wv   = threadIdx.x >> 5;
    const int bi   = blockIdx.x * WAVES_PER_BLOCK + wv;
    if (bi < batch) {
        const int idx = elt[bi];                 // wave-uniform
        const float* __restrict__ col = W + (u64)(u32)idx;
        const int e = lane << 2;
        const u64 vs = (u64)(u32)vocab;
        // 4 loads off one 64-bit base; byte offsets {0,1,2,3}*V*4 differ by
        // 400000B -> representable as 24-bit immediate offsets.
        const float* base = col + (u64)e * vs;
        float x = base[0];
        float y = base[vs];
        float z = base[2 * vs];
        float w = base[3 * vs];

        const float4 bb = *reinterpret_cast<const float4*>(&sbias[e]);  // ds_load_b128
        float4 r;
        r.x = x + bb.x;
        r.y = y + bb.y;
        r.z = z + bb.z;
        r.w = w + bb.w;
        // Wave's 32 lanes cover one contiguous 512B output row: b128 stores.
        *reinterpret_cast<float4*>(&out[(u64)bi * EMB + e]) = r;
    }
}

extern "C" void kernel_launch(void* const* d_in, const int* in_sizes, int n_in,
                              void* d_out, int out_size, void* d_ws, size_t ws_size,
                              hipStream_t stream) {
    const int*   elt = (const int*)d_in[0];
    const float* W   = (const float*)d_in[1];
    const float* b   = (const float*)d_in[2];
    float*       out = (float*)d_out;

    const int batch = in_sizes[0];
    const int vocab = in_sizes[1] / EMB;

    const int blocks = (batch + WAVES_PER_BLOCK - 1) / WAVES_PER_BLOCK;
    hipLaunchKernelGGL(emb_gather_kernel, dim3(blocks), dim3(THREADS), 0, stream,
                       elt, W, b, out, batch, vocab);
}